// L1LossWithPenalty_63333587746914
// MI455X (gfx1250) — compile-verified
//
#include <hip/hip_runtime.h>
#include <cstdint>

// Problem constants (from the reference)
#define NROWS 1048576
#define NC 27
#define BLOCK 128
#define WAVES_PER_BLOCK (BLOCK / 32)    // 4
#define GRID 1024
#define TILE_ROWS 32                    // rows staged per wave per tile
#define TILE_DW (TILE_ROWS * NC)        // 864 dwords
#define TILE_BYTES (TILE_DW * 4)        // 3456 bytes
// async instructions per tile: 2 arrays x (6 x b128 + 3 x b32) = 18
// -> with two tiles in flight, s_wait_asynccnt 18 retires the older tile
//    (ASYNCcnt completes in order).

// ---------------------------------------------------------------------------
// Async global -> LDS copy of one array's tile slice (CDNA5 ASYNCcnt path).
// The 24-bit instruction offset is added to BOTH the global and the LDS
// address (CDNA5 ISA), so one base VGPR per stream plus
// immediate offsets walks the tile in lockstep.
// Layout: 6 x b128 cover dwords [0,768); 3 x b32 cover dwords [768,864).
// ---------------------------------------------------------------------------
__device__ __forceinline__ void async_copy_stream(uint32_t lds_base,
                                                  const float* __restrict__ gbase,
                                                  uint32_t tile_byte,
                                                  uint32_t lane) {
    const uint64_t sbase = (uint64_t)(uintptr_t)gbase;

    uint32_t g128 = tile_byte + lane * 16u;
    uint32_t l128 = lds_base + lane * 16u;
    asm volatile(
        "global_load_async_to_lds_b128 %0, %1, %2 offset:0\n\t"
        "global_load_async_to_lds_b128 %0, %1, %2 offset:512\n\t"
        "global_load_async_to_lds_b128 %0, %1, %2 offset:1024\n\t"
        "global_load_async_to_lds_b128 %0, %1, %2 offset:1536\n\t"
        "global_load_async_to_lds_b128 %0, %1, %2 offset:2048\n\t"
        "global_load_async_to_lds_b128 %0, %1, %2 offset:2560\n\t"
        :: "v"(l128), "v"(g128), "s"(sbase) : "memory");

    uint32_t g32 = tile_byte + 3072u + lane * 4u;
    uint32_t l32 = lds_base + 3072u + lane * 4u;
    asm volatile(
        "global_load_async_to_lds_b32 %0, %1, %2 offset:0\n\t"
        "global_load_async_to_lds_b32 %0, %1, %2 offset:128\n\t"
        "global_load_async_to_lds_b32 %0, %1, %2 offset:256\n\t"
        :: "v"(l32), "v"(g32), "s"(sbase) : "memory");
}

// WAR guard: the DMA engine must not overwrite a staging buffer while a
// previous compute pass still has ds_loads in flight from it.
__device__ __forceinline__ void wait_ds_zero() {
    asm volatile("s_wait_dscnt 0" ::: "memory");
}
__device__ __forceinline__ void wait_async_le18() {
    asm volatile("s_wait_asynccnt 18" ::: "memory");
}
__device__ __forceinline__ void wait_async_zero() {
    asm volatile("s_wait_asynccnt 0" ::: "memory");
}

// Per-row payload: L1 sum + first-occurrence argmax of both vectors,
// then penalty gather and weighted accumulate.
__device__ __forceinline__ float row_contrib(const float* __restrict__ rp,
                                             const float* __restrict__ rt,
                                             const float* __restrict__ pen_s) {
    float s = 0.0f;
    float pmax = -__builtin_inff(); int pidx = 0;
    float tmax = -__builtin_inff(); int tidx = 0;
#pragma unroll
    for (int c = 0; c < NC; ++c) {
        float p = rp[c];
        float t = rt[c];
        s += __builtin_fabsf(p - t);
        if (p > pmax) { pmax = p; pidx = c; }   // strict > : first max wins
        if (t > tmax) { tmax = t; tidx = c; }
    }
    return s * pen_s[pidx * NC + tidx];
}

__global__ __launch_bounds__(BLOCK) void l1pen_main_kernel(
        const float* __restrict__ pred,
        const float* __restrict__ tgt,
        const float* __restrict__ pen,
        float* __restrict__ partial) {
    __shared__ float s_pen[NC * NC];                          //  2916 B
    __shared__ float s_pred[WAVES_PER_BLOCK][2][TILE_DW];     // 27648 B
    __shared__ float s_tgt [WAVES_PER_BLOCK][2][TILE_DW];     // 27648 B
    __shared__ float s_red[BLOCK];                            //   512 B
                                                              // ~58.7 KB total
    const uint32_t tid  = threadIdx.x;
    const uint32_t lane = tid & 31u;
    const uint32_t wave = tid >> 5;

    // Cooperative penalty-table load (read-hot, 729 floats).
    for (int i = (int)tid; i < NC * NC; i += BLOCK) s_pen[i] = pen[i];
    __syncthreads();

    // Generic pointers to __shared__ carry the LDS byte offset in their low
    // 32 bits (flat->LDS mapping truncates to addr[31:0]).
    const uint32_t lds_pred[2] = {
        (uint32_t)(uintptr_t)(&s_pred[wave][0][0]),
        (uint32_t)(uintptr_t)(&s_pred[wave][1][0]) };
    const uint32_t lds_tgt[2] = {
        (uint32_t)(uintptr_t)(&s_tgt[wave][0][0]),
        (uint32_t)(uintptr_t)(&s_tgt[wave][1][0]) };

    const uint32_t gwave  = blockIdx.x * WAVES_PER_BLOCK + wave;
    const uint32_t nwaves = gridDim.x * WAVES_PER_BLOCK;      // 4096
    const int64_t  stride = (int64_t)nwaves * TILE_ROWS;      // 131072 rows

    float acc = 0.0f;

    int64_t row0 = (int64_t)gwave * TILE_ROWS;
    if (row0 + TILE_ROWS <= NROWS) {
        // ---- software-pipelined double buffer over full tiles ----
        uint32_t buf = 0;
        {   // prologue: stage tile 0 into buf 0
            const uint32_t tb = (uint32_t)(row0 * NC * 4);
            wait_ds_zero();
            async_copy_stream(lds_pred[buf], pred, tb, lane);
            async_copy_stream(lds_tgt[buf],  tgt,  tb, lane);
        }
        for (;;) {
            const int64_t next = row0 + stride;
            const bool have_next = (next + TILE_ROWS <= NROWS);  // wave-uniform
            if (have_next) {
                const uint32_t tb = (uint32_t)(next * NC * 4);
                wait_ds_zero();                       // WAR vs compute(k-1)
                async_copy_stream(lds_pred[buf ^ 1u], pred, tb, lane);
                async_copy_stream(lds_tgt[buf ^ 1u],  tgt,  tb, lane);
                wait_async_le18();                    // tile k resident
            } else {
                wait_async_zero();                    // drain last tile
            }
            // Row stride 27 dwords is odd -> 32 lanes hit 32 distinct banks.
            acc += row_contrib(&s_pred[wave][buf][lane * NC],
                               &s_tgt[wave][buf][lane * NC], s_pen);
            if (!have_next) { row0 = next; break; }
            row0 = next;
            buf ^= 1u;
        }
    }
    // Ragged tail (dead for these constants: 4096*32 divides 2^20 exactly).
    for (; row0 < NROWS; row0 += stride) {
        const int64_t row = row0 + lane;
        if (row < NROWS)
            acc += row_contrib(pred + row * NC, tgt + row * NC, s_pen);
    }

    // Deterministic block reduction (fixed tree order).
    s_red[tid] = acc;
    __syncthreads();
    for (int s = BLOCK / 2; s > 0; s >>= 1) {
        if ((int)tid < s) s_red[tid] += s_red[tid + s];
        __syncthreads();
    }
    if (tid == 0) partial[blockIdx.x] = s_red[0];
}

__global__ __launch_bounds__(256) void l1pen_reduce_kernel(
        const float* __restrict__ partial, float* __restrict__ out, int nblocks) {
    __shared__ float s[256];
    const int tid = threadIdx.x;
    float a = 0.0f;
    for (int i = tid; i < nblocks; i += 256) a += partial[i];
    s[tid] = a;
    __syncthreads();
    for (int k = 128; k > 0; k >>= 1) {
        if (tid < k) s[tid] += s[tid + k];
        __syncthreads();
    }
    // N = 2^20: the mean is an exact power-of-two scale.
    if (tid == 0) out[0] = s[0] * (1.0f / (float)NROWS);
}

extern "C" void kernel_launch(void* const* d_in, const int* in_sizes, int n_in,
                              void* d_out, int out_size, void* d_ws, size_t ws_size,
                              hipStream_t stream) {
    (void)in_sizes; (void)n_in; (void)out_size; (void)ws_size;
    const float* pred = (const float*)d_in[0];
    const float* tgt  = (const float*)d_in[1];
    const float* pen  = (const float*)d_in[2];
    float* out        = (float*)d_out;
    float* partials   = (float*)d_ws;     // GRID floats = 4 KB of scratch

    l1pen_main_kernel<<<GRID, BLOCK, 0, stream>>>(pred, tgt, pen, partials);
    l1pen_reduce_kernel<<<1, 256, 0, stream>>>(partials, out, GRID);
}